// BEVFormerEncoder_13606456393870
// MI455X (gfx1250) — compile-verified
//
#include <hip/hip_runtime.h>
#include <math.h>

// ---------------------------------------------------------------------------
// BEVFormer encoder for MI455X (gfx1250, wave32).
// All GEMMs run through V_WMMA_F32_16X16X4_F32 (fp32-exact, matrix pipe).
// Workspace usage ~ 42 MB of fp32 scratch in d_ws.
// ---------------------------------------------------------------------------

typedef __attribute__((ext_vector_type(2))) float v2f;
typedef __attribute__((ext_vector_type(8))) float v8f;

#define NQ     2500
#define DIM    256
#define NHEADS 8
#define HD     32
#define NCAMS  6
#define NPILH  4
#define NVAL   7979
#define BEVH   50
#define BEVW   50

// ---------------------------------------------------------------------------
// Generic fp32 GEMM:  C[M,N] = A[M,K] @ W[K,N] + bias[N]  (optional ReLU)
// One wave computes one 16x16 tile of C via V_WMMA_F32_16X16X4_F32.
// Block = 128 threads = 4 waves covering a 16x64 strip of C.
// N must be a multiple of 64 (all uses: 64,128,256,512). M is edge-guarded.
// ---------------------------------------------------------------------------
__global__ __launch_bounds__(128)
void gemm_wmma_f32(const float* __restrict__ A, const float* __restrict__ W,
                   const float* __restrict__ bias, float* __restrict__ C,
                   int M, int N, int K, int relu) {
  const int lane = threadIdx.x & 31;
  const int wave = threadIdx.x >> 5;
  const int m0   = blockIdx.x * 16;
  const int n0   = (blockIdx.y * 4 + wave) * 16;
  const int l15  = lane & 15;
  const int hsel = lane >> 4;          // 0: K pair {0,1}, 1: K pair {2,3}

  const int am    = m0 + l15;                       // A row this lane fetches
  const int amc   = (am < M) ? am : (M - 1);        // clamped (avoid OOB)
  const float avl = (am < M) ? 1.f : 0.f;
  const int bn    = n0 + l15;                       // B column this lane fetches

  v8f acc = {};
  for (int k = 0; k < K; k += 4) {
    const int kk = k + hsel * 2;
    v2f a, b;
    const float* ap = A + (size_t)amc * K + kk;     // contiguous pair -> b64 load
    a.x = ap[0] * avl;
    a.y = ap[1] * avl;
    b.x = W[(size_t)kk * N + bn];
    b.y = W[(size_t)(kk + 1) * N + bn];
    acc = __builtin_amdgcn_wmma_f32_16x16x4_f32(
        /*neg_a=*/false, a, /*neg_b=*/false, b,
        /*c_mod=*/(short)0, acc, /*reuse_a=*/false, /*reuse_b=*/false);
  }

  const int   cn = n0 + l15;
  const float bv = bias[cn];
#pragma unroll
  for (int r = 0; r < 8; ++r) {
    const int cm = m0 + r + hsel * 8;               // C/D layout: VGPR r = row r / r+8
    if (cm < M) {
      float v = acc[r] + bv;
      if (relu) v = fmaxf(v, 0.f);
      C[(size_t)cm * N + cn] = v;
    }
  }
}

// ---------------------------------------------------------------------------
// Bilinear sample (zero pad outside image), channel-interleaved value field.
// loc already in pixel coords: x = loc_norm*W - 0.5 computed by caller.
// ---------------------------------------------------------------------------
__device__ __forceinline__ float bilin(const float* __restrict__ vf, int chan,
                                       float x, float y, int H, int Wd) {
  const float x0 = floorf(x), y0 = floorf(y);
  const float wx = x - x0, wy = y - y0;
  float s = 0.f;
#pragma unroll
  for (int dy = 0; dy < 2; ++dy) {
#pragma unroll
    for (int dx = 0; dx < 2; ++dx) {
      const float xc = x0 + (float)dx, yc = y0 + (float)dy;
      if (xc >= 0.f && xc <= (float)(Wd - 1) && yc >= 0.f && yc <= (float)(H - 1)) {
        const float wgt = (dx ? wx : 1.f - wx) * (dy ? wy : 1.f - wy);
        s += wgt * vf[(size_t)((int)yc * Wd + (int)xc) * DIM + chan];
      }
    }
  }
  return s;
}

// ---------------------------------------------------------------------------
// Precompute: project 3D BEV reference points through lidar2img.
// Thread = (cam, query). Writes ref_cam[(c*NQ+q)*NPIL+p][2] and vis0[c*NQ+q].
// ---------------------------------------------------------------------------
__global__ void proj_kernel(const float* __restrict__ l2i,
                            float* __restrict__ ref_cam,
                            float* __restrict__ vis0) {
  const int i = blockIdx.x * blockDim.x + threadIdx.x;
  if (i >= NCAMS * NQ) return;
  const int c = i / NQ, q = i % NQ;
  const int hh = q / BEVW, ww = q % BEVW;
  const float xr = (ww + 0.5f) / (float)BEVW;
  const float yr = (hh + 0.5f) / (float)BEVH;
  const float X = xr * 102.4f - 51.2f;
  const float Y = yr * 102.4f - 51.2f;
  const float* M = l2i + c * 16;
  bool any = false;
#pragma unroll
  for (int p = 0; p < NPILH; ++p) {
    const float zr = (0.5f + (float)p * (7.0f / 3.0f)) / 8.0f; // linspace(.5,7.5,4)/8
    const float Z = zr * 8.0f - 5.0f;
    const float p0 = M[0] * X + M[1] * Y + M[2]  * Z + M[3];
    const float p1 = M[4] * X + M[5] * Y + M[6]  * Z + M[7];
    const float p2 = M[8] * X + M[9] * Y + M[10] * Z + M[11];
    const bool  mz = p2 > 1e-5f;
    const float zz = fmaxf(p2, 1e-5f);
    const float x = (p0 / zz) * (1.0f / 800.0f);
    const float y = (p1 / zz) * (1.0f / 480.0f);
    any = any || (mz && y > 0.f && y < 1.f && x > 0.f && x < 1.f);
    ref_cam[((size_t)i * NPILH + p) * 2 + 0] = x;
    ref_cam[((size_t)i * NPILH + p) * 2 + 1] = y;
  }
  vis0[i] = any ? 1.f : 0.f;
}

// count[q] = max(1, sum_c vis0);  hyb[0][q] = ref2d + shift, hyb[1][q] = ref2d
__global__ void prep2_kernel(const float* __restrict__ vis0,
                             const float* __restrict__ shift,
                             float* __restrict__ count,
                             float* __restrict__ hyb) {
  const int q = blockIdx.x * blockDim.x + threadIdx.x;
  if (q >= NQ) return;
  float s = 0.f;
#pragma unroll
  for (int c = 0; c < NCAMS; ++c) s += vis0[c * NQ + q];
  count[q] = fmaxf(s, 1.f);
  const float rx = ((q % BEVW) + 0.5f) / (float)BEVW;
  const float ry = ((q / BEVW) + 0.5f) / (float)BEVH;
  hyb[(size_t)q * 2 + 0] = rx + shift[0];
  hyb[(size_t)q * 2 + 1] = ry + shift[1];
  hyb[(size_t)(NQ + q) * 2 + 0] = rx;
  hyb[(size_t)(NQ + q) * 2 + 1] = ry;
}

// prev_stack = [prev_bev ; bev_query]  (fixed for the whole forward)
__global__ void build_stack(const float* __restrict__ prev_bev,
                            const float* __restrict__ bev_query,
                            float* __restrict__ ps) {
  const int i = blockIdx.x * blockDim.x + threadIdx.x;
  if (i >= NQ * DIM) return;
  ps[i] = prev_bev[i];
  ps[NQ * DIM + i] = bev_query[i];
}

// qc[q] = concat(prev_bev[q], q_cur[q] + bev_pos[q])
__global__ void build_qc(const float* __restrict__ prev_bev,
                         const float* __restrict__ qcur,
                         const float* __restrict__ pos,
                         float* __restrict__ qc) {
  const int i = blockIdx.x * blockDim.x + threadIdx.x;
  if (i >= NQ * DIM) return;
  const int q = i / DIM, t = i % DIM;
  qc[(size_t)q * 512 + t] = prev_bev[i];
  qc[(size_t)q * 512 + 256 + t] = qcur[i] + pos[i];
}

// in-place softmax over contiguous groups
__global__ void softmax_rows(float* __restrict__ a, int ngroups, int glen) {
  const int i = blockIdx.x * blockDim.x + threadIdx.x;
  if (i >= ngroups) return;
  float* p = a + (size_t)i * glen;
  float m = -1e30f;
  for (int j = 0; j < glen; ++j) m = fmaxf(m, p[j]);
  float s = 0.f;
  for (int j = 0; j < glen; ++j) { const float e = __expf(p[j] - m); p[j] = e; s += e; }
  const float inv = 1.f / s;
  for (int j = 0; j < glen; ++j) p[j] *= inv;
}

__global__ void zero_buf(float* __restrict__ a, int n) {
  const int i = blockIdx.x * blockDim.x + threadIdx.x;
  if (i < n) a[i] = 0.f;
}

// ---------------------------------------------------------------------------
// TSA deformable sampling on the single 50x50 level, hybrid batch (prev,cur),
// fused mean over the 2 hybrid levels. Thread = (q, head, chan).
// ---------------------------------------------------------------------------
__global__ void tsa_sample(const float* __restrict__ v,    // (2*NQ, 256)
                           const float* __restrict__ off,  // (NQ,128)
                           const float* __restrict__ aw,   // (NQ,64) softmaxed
                           const float* __restrict__ hyb,  // (2,NQ,2)
                           float* __restrict__ out) {      // (NQ,256)
  const int i = blockIdx.x * blockDim.x + threadIdx.x;
  if (i >= NQ * DIM) return;
  const int q = i / DIM, t = i % DIM, hh = t / HD;
  float acc = 0.f;
#pragma unroll
  for (int b2 = 0; b2 < 2; ++b2) {
    const float rx = hyb[(size_t)(b2 * NQ + q) * 2 + 0];
    const float ry = hyb[(size_t)(b2 * NQ + q) * 2 + 1];
    const float* vf = v + (size_t)b2 * NQ * DIM;
#pragma unroll
    for (int p = 0; p < 4; ++p) {
      const size_t ob = (size_t)q * 128 + hh * 16 + b2 * 8 + p * 2;
      const float lx = rx + off[ob + 0] * (1.f / BEVW);
      const float ly = ry + off[ob + 1] * (1.f / BEVH);
      const float w  = aw[(size_t)q * 64 + hh * 8 + b2 * 4 + p];
      acc += w * bilin(vf, t, lx * BEVW - 0.5f, ly * BEVH - 0.5f, BEVH, BEVW);
    }
  }
  out[i] = 0.5f * acc;  // mean over the two hybrid levels
}

// ---------------------------------------------------------------------------
// SCA deformable sampling for one camera over 4 pyramid levels (8 pts each),
// accumulating slots += attn_out * vis0[c,q] / count[q].
// ---------------------------------------------------------------------------
__global__ void sca_sample(const float* __restrict__ vcam,  // (NVAL,256)
                           const float* __restrict__ off,   // (NQ,512)
                           const float* __restrict__ aw,    // (NQ,256) softmaxed
                           const float* __restrict__ refc,  // (NCAMS,NQ,NPIL,2)
                           const float* __restrict__ vis0,  // (NCAMS,NQ)
                           const float* __restrict__ count, // (NQ)
                           float* __restrict__ slots,       // (NQ,256)
                           int cam) {
  const int i = blockIdx.x * blockDim.x + threadIdx.x;
  if (i >= NQ * DIM) return;
  const int q = i / DIM, t = i % DIM, hh = t / HD;
  const float vw = vis0[cam * NQ + q];
  if (vw == 0.f) return;

  const int HS[4] = {60, 30, 15, 8};
  const int WS[4] = {100, 50, 25, 13};
  const int SO[4] = {0, 6000, 7500, 7875};

  float acc = 0.f;
  for (int l = 0; l < 4; ++l) {
    const float invW = 1.f / (float)WS[l], invH = 1.f / (float)HS[l];
    const float* vf = vcam + (size_t)SO[l] * DIM;
#pragma unroll
    for (int p8 = 0; p8 < 8; ++p8) {
      const size_t rb = ((size_t)(cam * NQ + q) * NPILH + (p8 & 3)) * 2;
      const size_t ob = (size_t)q * 512 + hh * 64 + l * 16 + p8 * 2;
      const float lx = refc[rb + 0] + off[ob + 0] * invW;
      const float ly = refc[rb + 1] + off[ob + 1] * invH;
      const float w  = aw[(size_t)q * 256 + hh * 32 + l * 8 + p8];
      acc += w * bilin(vf, t, lx * WS[l] - 0.5f, ly * HS[l] - 0.5f, HS[l], WS[l]);
    }
  }
  slots[i] += acc * vw / count[q];
}

// ---------------------------------------------------------------------------
// out = LayerNorm(x + res) * g + b   (one block per query row, LDS reduce)
// ---------------------------------------------------------------------------
__global__ __launch_bounds__(256)
void add_ln(const float* __restrict__ x, const float* __restrict__ res,
            const float* __restrict__ g, const float* __restrict__ b,
            float* __restrict__ out) {
  __shared__ float sm[256];
  const int q = blockIdx.x, t = threadIdx.x;
  const float v = x[(size_t)q * 256 + t] + res[(size_t)q * 256 + t];
  sm[t] = v; __syncthreads();
  for (int s = 128; s > 0; s >>= 1) { if (t < s) sm[t] += sm[t + s]; __syncthreads(); }
  const float mean = sm[0] * (1.f / 256.f); __syncthreads();
  const float d = v - mean;
  sm[t] = d * d; __syncthreads();
  for (int s = 128; s > 0; s >>= 1) { if (t < s) sm[t] += sm[t + s]; __syncthreads(); }
  const float var = sm[0] * (1.f / 256.f);
  out[(size_t)q * 256 + t] = d * rsqrtf(var + 1e-5f) * g[t] + b[t];
}

// ---------------------------------------------------------------------------
static inline int cdiv(int a, int b) { return (a + b - 1) / b; }

extern "C" void kernel_launch(void* const* d_in, const int* in_sizes, int n_in,
                              void* d_out, int out_size, void* d_ws, size_t ws_size,
                              hipStream_t stream) {
  (void)in_sizes; (void)n_in; (void)out_size; (void)ws_size;
  const float* bev_query = (const float*)d_in[0];
  const float* bev_pos   = (const float*)d_in[1];
  const float* feat      = (const float*)d_in[2];   // (6, 7979, 1, 256)
  const float* prev_bev  = (const float*)d_in[3];
  const float* lidar2img = (const float*)d_in[4];
  const float* shift     = (const float*)d_in[5];
  const float* tsa_off_w = (const float*)d_in[6];
  const float* tsa_off_b = (const float*)d_in[7];
  const float* tsa_aw_w  = (const float*)d_in[8];
  const float* tsa_aw_b  = (const float*)d_in[9];
  const float* tsa_val_w = (const float*)d_in[10];
  const float* tsa_val_b = (const float*)d_in[11];
  const float* tsa_out_w = (const float*)d_in[12];
  const float* tsa_out_b = (const float*)d_in[13];
  const float* sca_val_w = (const float*)d_in[14];
  const float* sca_val_b = (const float*)d_in[15];
  const float* sca_off_w = (const float*)d_in[16];
  const float* sca_off_b = (const float*)d_in[17];
  const float* sca_aw_w  = (const float*)d_in[18];
  const float* sca_aw_b  = (const float*)d_in[19];
  const float* sca_out_w = (const float*)d_in[20];
  const float* sca_out_b = (const float*)d_in[21];
  const float* ffn_w1    = (const float*)d_in[22];
  const float* ffn_b1    = (const float*)d_in[23];
  const float* ffn_w2    = (const float*)d_in[24];
  const float* ffn_b2    = (const float*)d_in[25];
  const float* ln_g      = (const float*)d_in[26];
  const float* ln_b      = (const float*)d_in[27];

  float* q = (float*)d_out;                 // current query lives in d_out
  float* ws = (float*)d_ws;

  // workspace layout (fp32 elements)
  float* ref_cam = ws;                      // 120000
  float* vis0    = ws + 120000;             // 15000
  float* count   = ws + 135000;             // 2500
  float* hyb     = ws + 137500;             // 10000
  float* pstack  = ws + 147500;             // 2*NQ*DIM = 1,280,000
  float* qc      = ws + 1427500;            // NQ*512  (reused as FFN hidden)
  float* v_tsa   = ws + 2707500;            // 2*NQ*DIM
  float* off_tsa = ws + 3987500;            // NQ*128
  float* aw_tsa  = ws + 4307500;            // NQ*64
  float* off_sca = ws + 4467500;            // NQ*512
  float* aw_sca  = ws + 5747500;            // NQ*256
  float* v_cam   = ws + 6387500;            // NVAL*DIM = 2,042,624
  float* slots   = ws + 8430124;            // NQ*DIM
  float* tmp2    = ws + 9070124;            // NQ*DIM
  float* tmp1    = ws + 9710124;            // NQ*DIM   (total ~10.35M floats)

  const dim3 eb(256);
  #define EG(n) dim3(cdiv((n), 256))

  // ---- init: q = bev_query, precompute projections / refs / prev stack ----
  hipMemcpyAsync(q, bev_query, (size_t)NQ * DIM * sizeof(float),
                 hipMemcpyDeviceToDevice, stream);
  proj_kernel<<<EG(NCAMS * NQ), eb, 0, stream>>>(lidar2img, ref_cam, vis0);
  prep2_kernel<<<EG(NQ), eb, 0, stream>>>(vis0, shift, count, hyb);
  build_stack<<<EG(NQ * DIM), eb, 0, stream>>>(prev_bev, bev_query, pstack);

  for (int i = 0; i < 2; ++i) {
    // ================= Temporal self-attention =================
    build_qc<<<EG(NQ * DIM), eb, 0, stream>>>(prev_bev, q, bev_pos, qc);
    // value projection over the hybrid stack: (5000,256)@(256,256)
    gemm_wmma_f32<<<dim3(cdiv(2 * NQ, 16), 4), 128, 0, stream>>>(
        pstack, tsa_val_w + (size_t)i * 256 * 256, tsa_val_b + i * 256,
        v_tsa, 2 * NQ, 256, 256, 0);
    // offsets: (2500,512)@(512,128)
    gemm_wmma_f32<<<dim3(cdiv(NQ, 16), 2), 128, 0, stream>>>(
        qc, tsa_off_w + (size_t)i * 512 * 128, tsa_off_b + i * 128,
        off_tsa, NQ, 128, 512, 0);
    // attention weights: (2500,512)@(512,64) then softmax over 4
    gemm_wmma_f32<<<dim3(cdiv(NQ, 16), 1), 128, 0, stream>>>(
        qc, tsa_aw_w + (size_t)i * 512 * 64, tsa_aw_b + i * 64,
        aw_tsa, NQ, 64, 512, 0);
    softmax_rows<<<EG(NQ * 16), eb, 0, stream>>>(aw_tsa, NQ * 16, 4);
    tsa_sample<<<EG(NQ * DIM), eb, 0, stream>>>(v_tsa, off_tsa, aw_tsa, hyb, tmp1);
    gemm_wmma_f32<<<dim3(cdiv(NQ, 16), 4), 128, 0, stream>>>(
        tmp1, tsa_out_w + (size_t)i * 256 * 256, tsa_out_b + i * 256,
        tmp2, NQ, 256, 256, 0);
    add_ln<<<NQ, 256, 0, stream>>>(tmp2, q, ln_g + (size_t)(i * 3 + 0) * 256,
                                   ln_b + (size_t)(i * 3 + 0) * 256, q);

    // ================= Spatial cross-attention =================
    // qcam is a broadcast of q over cameras -> off/aw are camera-independent
    gemm_wmma_f32<<<dim3(cdiv(NQ, 16), 8), 128, 0, stream>>>(
        q, sca_off_w + (size_t)i * 256 * 512, sca_off_b + i * 512,
        off_sca, NQ, 512, 256, 0);
    gemm_wmma_f32<<<dim3(cdiv(NQ, 16), 4), 128, 0, stream>>>(
        q, sca_aw_w + (size_t)i * 256 * 256, sca_aw_b + i * 256,
        aw_sca, NQ, 256, 256, 0);
    softmax_rows<<<EG(NQ * 8), eb, 0, stream>>>(aw_sca, NQ * 8, 32);
    zero_buf<<<EG(NQ * DIM), eb, 0, stream>>>(slots, NQ * DIM);
    for (int c = 0; c < NCAMS; ++c) {
      // per-cam value projection: (7979,256)@(256,256)
      gemm_wmma_f32<<<dim3(cdiv(NVAL, 16), 4), 128, 0, stream>>>(
          feat + (size_t)c * NVAL * DIM,
          sca_val_w + (size_t)i * 256 * 256, sca_val_b + i * 256,
          v_cam, NVAL, 256, 256, 0);
      sca_sample<<<EG(NQ * DIM), eb, 0, stream>>>(
          v_cam, off_sca, aw_sca, ref_cam, vis0, count, slots, c);
    }
    gemm_wmma_f32<<<dim3(cdiv(NQ, 16), 4), 128, 0, stream>>>(
        slots, sca_out_w + (size_t)i * 256 * 256, sca_out_b + i * 256,
        tmp2, NQ, 256, 256, 0);
    add_ln<<<NQ, 256, 0, stream>>>(tmp2, q, ln_g + (size_t)(i * 3 + 1) * 256,
                                   ln_b + (size_t)(i * 3 + 1) * 256, q);

    // ================= FFN =================
    gemm_wmma_f32<<<dim3(cdiv(NQ, 16), 8), 128, 0, stream>>>(
        q, ffn_w1 + (size_t)i * 256 * 512, ffn_b1 + i * 512,
        qc /* reuse as hidden */, NQ, 512, 256, 1);
    gemm_wmma_f32<<<dim3(cdiv(NQ, 16), 4), 128, 0, stream>>>(
        qc, ffn_w2 + (size_t)i * 512 * 256, ffn_b2 + i * 256,
        tmp2, NQ, 256, 512, 0);
    add_ln<<<NQ, 256, 0, stream>>>(tmp2, q, ln_g + (size_t)(i * 3 + 2) * 256,
                                   ln_b + (size_t)(i * 3 + 2) * 256, q);
  }
}